// CenterLossB_51951924413099
// MI455X (gfx1250) — compile-verified
//
#include <hip/hip_runtime.h>

#define D_DIM 512
#define NCLS 3

typedef __attribute__((ext_vector_type(2))) float v2f;
typedef __attribute__((ext_vector_type(8))) float v8f;

// ---------------------------------------------------------------------------
// Kernel 0: per-call init. Computes n_k = ||c_k||^2 (3 waves, one per class)
// and zeroes the two f64 global accumulators. Must run every call since the
// harness does not re-poison/restore d_ws between replays.
// ---------------------------------------------------------------------------
__global__ __launch_bounds__(96) void cl_setup(const float* __restrict__ centers,
                                               double* __restrict__ acc,
                                               float* __restrict__ cn) {
    int k = threadIdx.x >> 5;   // class, one wave32 per class
    int l = threadIdx.x & 31;
    const float* c = centers + k * D_DIM;
    float s = 0.f;
    for (int j = l; j < D_DIM; j += 32) { float v = c[j]; s = fmaf(v, v, s); }
    for (int off = 16; off > 0; off >>= 1) s += __shfl_down(s, off, 32);
    if (l == 0) cn[k] = s;
    if (threadIdx.x == 0) { acc[0] = 0.0; acc[1] = 0.0; }
}

// ---------------------------------------------------------------------------
// Kernel 1: streaming pass. One wave per 16-row tile of feat.
// D = A(16x4 feat slice) x B(4x16: centers^T padded with zeros) accumulated
// over 128 K-chunks via v_wmma_f32_16x16x4_f32. ||f||^2 via VALU fma on the
// same registers. Per-row own/dist combined, wave-reduced, block-reduced in
// LDS, then 2 f64 global atomics per block.
// ---------------------------------------------------------------------------
__global__ __launch_bounds__(256) void cl_main(const float* __restrict__ feat,
                                               const int*   __restrict__ label,
                                               const float* __restrict__ wei,
                                               const float* __restrict__ centers,
                                               const float* __restrict__ cn,
                                               double* __restrict__ acc,
                                               int nrows) {
    // B-operand staged per K-chunk in WMMA lane order:
    // chunk kk, lane ln reads float2 at centB + kk*64 + ln*2:
    //   .x = B[2*(ln>>4)+0][ln&15], .y = B[2*(ln>>4)+1][ln&15]
    __shared__ float centB[128 * 64];            // 32 KiB
    __shared__ float tileD[8][16 * 16];          // 8 KiB  (D spill per wave)
    __shared__ float sqv[8][32];                 // 1 KiB  (||f||^2 halves)
    __shared__ float cnS[4];
    __shared__ float redOwn, redDist;

    const int tid = threadIdx.x;

    for (int j = tid; j < 128 * 64; j += 256) {
        int kk  = j >> 6;
        int t   = j & 63;
        int lnn = t >> 1;          // target lane
        int h   = t & 1;           // which of the 2 B VGPRs
        int k   = 4 * kk + 2 * (lnn >> 4) + h;   // K row within tensor dim
        int n   = lnn & 15;        // N column (class)
        centB[j] = (n < NCLS) ? centers[n * D_DIM + k] : 0.0f;
    }
    if (tid == 0) {
        float c0 = cn[0], c1 = cn[1], c2 = cn[2];
        cnS[0] = c0; cnS[1] = c1; cnS[2] = c2; cnS[3] = c0 + c1 + c2;
        redOwn = 0.f; redDist = 0.f;
    }
    __syncthreads();

    const int  wid  = tid >> 5;
    const int  ln   = tid & 31;
    const int  half = ln >> 4;       // which K pair this lane holds in A
    const int  mcol = ln & 15;       // A row index / D column index
    const long base = ((long)blockIdx.x * 8 + wid) * 16;
    const bool valid = (base + 16) <= (long)nrows;

    v8f acc0 = {}; v8f acc1 = {};
    float sq = 0.f;

    if (valid) {
        const float* aptr = feat + (base + mcol) * (long)D_DIM + 2 * half;
        const float* bptr = centB + ln * 2;
        #pragma unroll 4
        for (int kk = 0; kk < 128; kk += 2) {
            v2f a0 = *(const v2f*)(aptr + 4 * kk);
            v2f a1 = *(const v2f*)(aptr + 4 * kk + 4);
            v2f b0 = *(const v2f*)(bptr + 64 * kk);
            v2f b1 = *(const v2f*)(bptr + 64 * kk + 64);
            acc0 = __builtin_amdgcn_wmma_f32_16x16x4_f32(false, a0, false, b0,
                                                         (short)0, acc0, false, false);
            acc1 = __builtin_amdgcn_wmma_f32_16x16x4_f32(false, a1, false, b1,
                                                         (short)0, acc1, false, false);
            sq = fmaf(a0.x, a0.x, sq); sq = fmaf(a0.y, a0.y, sq);
            sq = fmaf(a1.x, a1.x, sq); sq = fmaf(a1.y, a1.y, sq);
        }
        float* myTile = tileD[wid];
        #pragma unroll
        for (int r = 0; r < 8; ++r)           // D: VGPR r, lane ln -> M=r+8*half, N=ln&15
            myTile[(r + 8 * half) * 16 + mcol] = acc0[r] + acc1[r];
        sqv[wid][ln] = sq;
    }
    __syncthreads();

    float ownP = 0.f, distP = 0.f;
    if (valid && ln < 16) {
        const float* trow = tileD[wid] + ln * 16;
        float d0 = trow[0], d1 = trow[1], d2 = trow[2];
        float s  = sqv[wid][ln] + sqv[wid][ln + 16];
        long  row = base + ln;
        float w   = wei[row];
        int   lab = label[row];
        float dl  = (lab == 0) ? d0 : ((lab == 1) ? d1 : d2);
        float own = w * (s - 2.f * dl + cnS[lab]);
        float tot = w * (3.f * s - 2.f * (d0 + d1 + d2) + cnS[3]);
        ownP  = own;
        distP = tot - own;
    }
    for (int off = 16; off > 0; off >>= 1) {
        ownP  += __shfl_down(ownP,  off, 32);
        distP += __shfl_down(distP, off, 32);
    }
    if (ln == 0) { atomicAdd(&redOwn, ownP); atomicAdd(&redDist, distP); }
    __syncthreads();
    if (tid == 0) {
        atomicAdd(&acc[0], (double)redOwn);
        atomicAdd(&acc[1], (double)redDist);
    }
}

// ---------------------------------------------------------------------------
// Kernel 2: finalize scalar.
// ---------------------------------------------------------------------------
__global__ void cl_fin(const double* __restrict__ acc,
                       const int* __restrict__ bsz,
                       float* __restrict__ out) {
    if (threadIdx.x == 0 && blockIdx.x == 0) {
        double own = acc[0], dist = acc[1];
        out[0] = (float)(own * (1.0 + 1.0 / dist) * 0.5 / (double)bsz[0]);
    }
}

extern "C" void kernel_launch(void* const* d_in, const int* in_sizes, int n_in,
                              void* d_out, int out_size, void* d_ws, size_t ws_size,
                              hipStream_t stream) {
    const float* feat    = (const float*)d_in[0];
    const int*   label   = (const int*)  d_in[1];
    const float* wei     = (const float*)d_in[2];
    const float* centers = (const float*)d_in[3];
    const int*   bsz     = (const int*)  d_in[4];

    const int B = in_sizes[2];              // batch size = #weights

    double* acc = (double*)d_ws;            // [0]=own, [1]=dist (f64 accumulators)
    float*  cn  = (float*)((char*)d_ws + 2 * sizeof(double)); // n_k, 3 floats

    cl_setup<<<1, 96, 0, stream>>>(centers, acc, cn);

    int tiles  = B / 16;                    // 16 rows per wave
    int blocks = (tiles + 7) / 8;           // 8 waves per block
    cl_main<<<blocks, 256, 0, stream>>>(feat, label, wei, centers, cn, acc, B);

    cl_fin<<<1, 1, 0, stream>>>(acc, bsz, (float*)d_out);
}